// LORAMultiheadAttention_65309272703545
// MI455X (gfx1250) — compile-verified
//
#include <hip/hip_runtime.h>
#include <hip/hip_bf16.h>

// ---------------------------------------------------------------------------
// LoRA Multihead Attention forward for MI455X (gfx1250), fp32 via
// V_WMMA_F32_16X16X4_F32 (exact-precision matrix path).
//
//   k0: XA   = X @ lora_a.T            [4096,16]   (WMMA, K=1024)
//   k1: P    = X @ W.T + b (+ XA@LB.T) [4096,1024] (WMMA; A tile staged in LDS
//        via global_load_async_to_lds_b128, 32x64 output per wave)
//   k2: attention per (b, 16-row tile), loop 16 heads:
//        S = scale*Q@K.T + mask -> LDS (16x2052 padded, bank-conflict-free)
//        softmax in LDS; a_avg += P/16 (exclusive owner, no atomics)
//        O = P @ V (A-operand from LDS)             (WMMA, K=2048 split /2)
//   k3: out  = attn @ o_w.T + o_b                   (same as k1)
//
// Workspace: 4x [4096,1024] f32 + 2x [4096,16] f32 ~= 67.6 MB.
// ---------------------------------------------------------------------------

typedef float v2f __attribute__((ext_vector_type(2)));
typedef float v8f __attribute__((ext_vector_type(8)));

#define EMBED     1024
#define HEADS     16
#define HEAD_DIM  64
#define LORA_R    16
#define TGT       2048
#define SRC       2048
#define BSZ       2
#define MROWS     (TGT * BSZ)      // 4096
#define ATT_SCALE 0.125f           // 1/sqrt(64)
#define LORA_SCALE 1.0f            // alpha/r = 16/16

#define AP 1028                    // padded A-tile row stride (floats), 1028%64=4
#define SP 2052                    // padded S row stride (floats),      2052%64=4

__device__ __forceinline__ v8f wmma_f32(v2f a, v2f b, v8f c) {
  return __builtin_amdgcn_wmma_f32_16x16x4_f32(
      /*neg_a=*/false, a, /*neg_b=*/false, b,
      /*c_mod=*/(short)0, c, /*reuse_a=*/false, /*reuse_b=*/false);
}

__device__ __forceinline__ v2f ld2(const float* p) { return *(const v2f*)p; }

// Async copy 16B/lane (512B/wave) from global to LDS; tracked by ASYNCcnt.
// Operand order matches VGLOBAL asm: dst-LDS-offset VGPR, 64-bit vaddr, off.
__device__ __forceinline__ void async_copy_b128(unsigned int lds_byte_off,
                                                const float* g) {
  asm volatile("global_load_async_to_lds_b128 %0, %1, off"
               :: "v"(lds_byte_off), "v"(g)
               : "memory");
}
__device__ __forceinline__ void wait_async0() {
  asm volatile("s_wait_asynccnt 0x0" ::: "memory");
}

// ---------------------------------------------------------------------------
// k0: XA[m,r] = LORA_SCALE * sum_k X[m,k] * la[r,k]; one wave per 16-row tile.
// ---------------------------------------------------------------------------
__global__ void lora_xa_kernel(const float* __restrict__ X,   // [4096,1024]
                               const float* __restrict__ la,  // [16,1024]
                               float* __restrict__ XA)        // [4096,16]
{
  const int lane  = threadIdx.x & 31;
  const int w     = threadIdx.x >> 5;
  const int mBase = (blockIdx.x * 8 + w) * 16;
  const int hlf   = lane >> 4;
  const int lr    = lane & 15;

  v8f c = {};
  const float* xrow  = X  + (size_t)(mBase + lr) * EMBED + 2 * hlf;
  const float* larow = la + (size_t)lr * EMBED + 2 * hlf;
  for (int kb = 0; kb < EMBED; kb += 4)
    c = wmma_f32(ld2(xrow + kb), ld2(larow + kb), c);
#pragma unroll
  for (int i = 0; i < 8; ++i)
    XA[(size_t)(mBase + i + 8 * hlf) * LORA_R + lr] = c[i] * LORA_SCALE;
}

// ---------------------------------------------------------------------------
// k1/k3: P = X @ W.T + bias (+ XA @ LB.T).
// Block (8 waves) owns 32 rows x 512 cols; A tile [32 x 1024] async-staged in
// LDS (rows padded to AP=1028 -> conflict-free ds_load_b64 fragments).
// Each wave: 32x64 output = 8 accumulators; per k-step 2 LDS A + 4 global B
// fragment loads feed 8 WMMA. Weights are L2-resident (4MB vs 192MB L2).
// ---------------------------------------------------------------------------
__global__ void proj_gemm_kernel(const float* __restrict__ X,    // [4096,1024]
                                 const float* __restrict__ W,    // [1024,1024]
                                 const float* __restrict__ bias, // [1024]
                                 const float* __restrict__ XA,   // [4096,16] or null
                                 const float* __restrict__ LB,   // [1024,16] or null
                                 float* __restrict__ P)          // [4096,1024]
{
  extern __shared__ float sA[];               // [32][AP]
  const int tid   = threadIdx.x;
  const int lane  = tid & 31;
  const int w     = tid >> 5;
  const int hlf   = lane >> 4;
  const int lr    = lane & 15;
  const int mBase = blockIdx.x * 32;          // 0..4095 in steps of 32
  const int nBase = blockIdx.y * 512 + w * 64;

  // ---- stage A tile [32 x 1024] -> LDS via async b128 (32 instr/wave) ----
  {
    const unsigned int ldsBase = (unsigned int)__builtin_amdgcn_groupstaticsize();
    for (int j = 0; j < 32; ++j) {
      const int chunk = w * 32 + j;           // 0..255, 512B each
      const int row   = chunk >> 3;           // 8 chunks per 4KB row
      const int seg   = chunk & 7;
      const unsigned int ldsOff =
          ldsBase + (unsigned int)(row * (AP * 4) + seg * 512 + lane * 16);
      const float* g = X + (size_t)(mBase + row) * EMBED + seg * 128 + lane * 4;
      async_copy_b128(ldsOff, g);
    }
    wait_async0();
  }
  __syncthreads();

  v8f c00 = {}, c01 = {}, c02 = {}, c03 = {};
  v8f c10 = {}, c11 = {}, c12 = {}, c13 = {};
  const float* w0 = W + (size_t)(nBase +  0 + lr) * EMBED + 2 * hlf;
  const float* w1 = W + (size_t)(nBase + 16 + lr) * EMBED + 2 * hlf;
  const float* w2 = W + (size_t)(nBase + 32 + lr) * EMBED + 2 * hlf;
  const float* w3 = W + (size_t)(nBase + 48 + lr) * EMBED + 2 * hlf;
  const float* a0p = &sA[(     lr) * AP + 2 * hlf];
  const float* a1p = &sA[(16 + lr) * AP + 2 * hlf];

  for (int kb = 0; kb < EMBED; kb += 4) {
    __builtin_prefetch(w0 + kb + 256, 0, 1);   // global_prefetch_b8
    v2f a0 = ld2(a0p + kb);                    // ds_load_b64, conflict-free
    v2f a1 = ld2(a1p + kb);
    v2f b0 = ld2(w0 + kb);
    v2f b1 = ld2(w1 + kb);
    v2f b2 = ld2(w2 + kb);
    v2f b3 = ld2(w3 + kb);
    c00 = wmma_f32(a0, b0, c00);  c01 = wmma_f32(a0, b1, c01);
    c02 = wmma_f32(a0, b2, c02);  c03 = wmma_f32(a0, b3, c03);
    c10 = wmma_f32(a1, b0, c10);  c11 = wmma_f32(a1, b1, c11);
    c12 = wmma_f32(a1, b2, c12);  c13 = wmma_f32(a1, b3, c13);
  }

  if (XA != nullptr) {  // LoRA tail: K = 16 (XA already scaled by LORA_SCALE)
    const float* x0 = XA + (size_t)(mBase +      lr) * LORA_R + 2 * hlf;
    const float* x1 = XA + (size_t)(mBase + 16 + lr) * LORA_R + 2 * hlf;
    const float* l0 = LB + (size_t)(nBase +  0 + lr) * LORA_R + 2 * hlf;
    const float* l1 = LB + (size_t)(nBase + 16 + lr) * LORA_R + 2 * hlf;
    const float* l2 = LB + (size_t)(nBase + 32 + lr) * LORA_R + 2 * hlf;
    const float* l3 = LB + (size_t)(nBase + 48 + lr) * LORA_R + 2 * hlf;
#pragma unroll
    for (int kb = 0; kb < LORA_R; kb += 4) {
      v2f a0 = ld2(x0 + kb), a1 = ld2(x1 + kb);
      v2f b0 = ld2(l0 + kb), b1 = ld2(l1 + kb);
      v2f b2 = ld2(l2 + kb), b3 = ld2(l3 + kb);
      c00 = wmma_f32(a0, b0, c00);  c01 = wmma_f32(a0, b1, c01);
      c02 = wmma_f32(a0, b2, c02);  c03 = wmma_f32(a0, b3, c03);
      c10 = wmma_f32(a1, b0, c10);  c11 = wmma_f32(a1, b1, c11);
      c12 = wmma_f32(a1, b2, c12);  c13 = wmma_f32(a1, b3, c13);
    }
  }

  const float b0 = bias[nBase +  0 + lr];
  const float b1 = bias[nBase + 16 + lr];
  const float b2 = bias[nBase + 32 + lr];
  const float b3 = bias[nBase + 48 + lr];
#pragma unroll
  for (int i = 0; i < 8; ++i) {
    const int r0 = mBase + i + 8 * hlf;
    float* d0 = P + (size_t)r0 * EMBED + nBase + lr;
    d0[0]  = c00[i] + b0;  d0[16] = c01[i] + b1;
    d0[32] = c02[i] + b2;  d0[48] = c03[i] + b3;
    float* d1 = d0 + (size_t)16 * EMBED;
    d1[0]  = c10[i] + b0;  d1[16] = c11[i] + b1;
    d1[32] = c12[i] + b2;  d1[48] = c13[i] + b3;
  }
}

// ---------------------------------------------------------------------------
// k2: attention. One workgroup (8 wave32) owns (b, 16-row t-tile), loops over
// 16 heads. S block 16 x 2052 (padded) in LDS; P-V A-fragments hit all 64
// banks conflict-free. a_avg ownership exclusive -> plain RMW.
// ---------------------------------------------------------------------------
__global__ void attn_kernel(const float* __restrict__ qp,     // [4096,1024]
                            const float* __restrict__ kp,     // [4096,1024]
                            const float* __restrict__ vp,     // [4096,1024]
                            const float* __restrict__ mask,   // [tgt,src]
                            float* __restrict__ attn_out,     // [4096,1024]
                            float* __restrict__ a_avg)        // [bsz,tgt,src]
{
  extern __shared__ float smem[];
  float* S   = smem;               // 16 x SP scores/probs
  float* red = smem + 16 * SP;     // 16 x 16 reductions
  float* Op  = red + 256;          // 2 x 16 x 64 partial O

  const int tid   = threadIdx.x;
  const int lane  = tid & 31;
  const int w     = tid >> 5;
  const int hlf   = lane >> 4;
  const int lr    = lane & 15;
  const int tBase = blockIdx.x * 16;
  const int b     = blockIdx.y;

  for (int h = 0; h < HEADS; ++h) {
    const int off = b * EMBED + h * HEAD_DIM;

    // ---- scores: S = scale * Q K^T + mask --------------------------------
    v2f aq[16];
    {
      const float* qrow = qp + (size_t)(tBase + lr) * (BSZ * EMBED) + off + 2 * hlf;
#pragma unroll
      for (int kk = 0; kk < 16; ++kk) aq[kk] = ld2(qrow + 4 * kk);
    }
    for (int it = 0; it < 16; ++it) {          // 8 waves x 16 = 128 s-tiles
      const int sBase = (w * 16 + it) * 16;
      const float* krow = kp + (size_t)(sBase + lr) * (BSZ * EMBED) + off + 2 * hlf;
      v8f c = {};
#pragma unroll
      for (int kk = 0; kk < 16; ++kk)
        c = wmma_f32(aq[kk], ld2(krow + 4 * kk), c);
#pragma unroll
      for (int i = 0; i < 8; ++i) {
        const int row = i + 8 * hlf;
        const int col = sBase + lr;
        S[row * SP + col] = c[i] * ATT_SCALE + mask[(size_t)(tBase + row) * SRC + col];
      }
    }
    __syncthreads();

    // ---- softmax per row (16 threads x 128 cols) -------------------------
    {
      const int row = tid >> 4;
      const int c16 = tid & 15;
      float* Srow = S + row * SP;
      const int c0 = c16 * 128;
      float m = -1e30f;
      for (int c = 0; c < 128; ++c) m = fmaxf(m, Srow[c0 + c]);
      red[row * 16 + c16] = m;
      __syncthreads();
      float rowmax = -1e30f;
#pragma unroll
      for (int i = 0; i < 16; ++i) rowmax = fmaxf(rowmax, red[row * 16 + i]);
      __syncthreads();
      float lsum = 0.f;
      for (int c = 0; c < 128; ++c) {
        const float e = __expf(Srow[c0 + c] - rowmax);
        Srow[c0 + c] = e;
        lsum += e;
      }
      red[row * 16 + c16] = lsum;
      __syncthreads();
      float rowsum = 0.f;
#pragma unroll
      for (int i = 0; i < 16; ++i) rowsum += red[row * 16 + i];
      const float inv   = 1.0f / rowsum;
      const float inv16 = 1.0f / (float)HEADS;
      float* g = a_avg + ((size_t)b * TGT + tBase + row) * SRC + c0;
      for (int c = 0; c < 128; ++c) {
        const float p = Srow[c0 + c] * inv;
        Srow[c0 + c] = p;
        if (h == 0) g[c]  = p * inv16;   // first head initializes (d_out poisoned)
        else        g[c] += p * inv16;   // exclusive owner -> no atomics
      }
    }
    __syncthreads();

    // ---- O = P @ V : 16x64, K=2048 split across 2 wave groups ------------
    {
      const int nt    = w & 3;
      const int kh    = w >> 2;
      const int dBase = nt * 16;
      const float* vcol = vp + off + dBase + lr;   // lanes -> coalesced 64B rows
      v8f c = {};
      const int k0 = kh * 1024;
      for (int sb = k0; sb < k0 + 1024; sb += 4) {
        v2f a = *(const v2f*)&S[lr * SP + sb + 2 * hlf];   // conflict-free b64
        v2f bf;
        bf.x = vcol[(size_t)(sb + 2 * hlf    ) * (BSZ * EMBED)];
        bf.y = vcol[(size_t)(sb + 2 * hlf + 1) * (BSZ * EMBED)];
        c = wmma_f32(a, bf, c);
      }
#pragma unroll
      for (int i = 0; i < 8; ++i)
        Op[kh * 1024 + (i + 8 * hlf) * 64 + dBase + lr] = c[i];
    }
    __syncthreads();

    // ---- combine K-halves, store head output -----------------------------
    for (int e = tid; e < 16 * HEAD_DIM; e += 256) {
      const int m = e >> 6, d = e & 63;
      const float val = Op[m * 64 + d] + Op[1024 + m * 64 + d];
      attn_out[((size_t)(tBase + m) * BSZ + b) * EMBED + h * HEAD_DIM + d] = val;
    }
    __syncthreads();
  }
}

// ---------------------------------------------------------------------------
extern "C" void kernel_launch(void* const* d_in, const int* in_sizes, int n_in,
                              void* d_out, int out_size, void* d_ws, size_t ws_size,
                              hipStream_t stream) {
  const float* q    = (const float*)d_in[0];
  const float* k    = (const float*)d_in[1];
  const float* v    = (const float*)d_in[2];
  const float* mask = (const float*)d_in[3];
  const float* q_w  = (const float*)d_in[4];
  const float* q_b  = (const float*)d_in[5];
  const float* lqa  = (const float*)d_in[6];
  const float* lqb  = (const float*)d_in[7];
  const float* k_w  = (const float*)d_in[8];
  const float* k_b  = (const float*)d_in[9];
  const float* v_w  = (const float*)d_in[10];
  const float* v_b  = (const float*)d_in[11];
  const float* lva  = (const float*)d_in[12];
  const float* lvb  = (const float*)d_in[13];
  const float* o_w  = (const float*)d_in[14];
  const float* o_b  = (const float*)d_in[15];

  float* out_o = (float*)d_out;                            // [2048,2,1024]
  float* a_avg = out_o + (size_t)MROWS * EMBED;            // [2,2048,2048]

  float* ws   = (float*)d_ws;
  float* qp   = ws;                                        // [4096,1024]
  float* kp   = qp   + (size_t)MROWS * EMBED;
  float* vp   = kp   + (size_t)MROWS * EMBED;
  float* attn = vp   + (size_t)MROWS * EMBED;
  float* xaq  = attn + (size_t)MROWS * EMBED;              // [4096,16]
  float* xav  = xaq  + (size_t)MROWS * LORA_R;             // [4096,16]

  const dim3 blk(256);

  // LoRA intermediates
  lora_xa_kernel<<<32, blk, 0, stream>>>(q, lqa, xaq);
  lora_xa_kernel<<<32, blk, 0, stream>>>(v, lva, xav);

  // Q/K/V projections: grid = (M/32, N/512), LDS = 32 x AP floats
  const size_t gemm_smem = (size_t)32 * AP * sizeof(float);  // 131584 B
  const dim3 ggrid(MROWS / 32, EMBED / 512);
  proj_gemm_kernel<<<ggrid, blk, gemm_smem, stream>>>(q, q_w, q_b, xaq, lqb, qp);
  proj_gemm_kernel<<<ggrid, blk, gemm_smem, stream>>>(k, k_w, k_b, nullptr, nullptr, kp);
  proj_gemm_kernel<<<ggrid, blk, gemm_smem, stream>>>(v, v_w, v_b, xav, lvb, vp);

  // Attention (LDS: 16 x SP scores + reductions + partial O)
  const size_t attn_smem = (size_t)(16 * SP + 256 + 2 * 16 * 64) * sizeof(float);
  attn_kernel<<<dim3(TGT / 16, BSZ), blk, attn_smem, stream>>>(
      qp, kp, vp, mask, attn, a_avg);

  // Output projection
  proj_gemm_kernel<<<ggrid, blk, gemm_smem, stream>>>(attn, o_w, o_b,
                                                      nullptr, nullptr, out_o);
}